// VectorQuantizer_75445395522048
// MI455X (gfx1250) — compile-verified
//
#include <hip/hip_runtime.h>

#define B_TOT   32768
#define D_DIM   256
#define K_CB    16
#define ROWS    64                 // rows of z per block
#define STAGE_R 4                  // rows per async stage (4*256*4B = 4KB = 256 lanes * 16B)
#define NSTAGE  (ROWS / STAGE_R)   // 16
#define NBLOCKS (B_TOT / ROWS)     // 512

// CDNA5 async global->LDS path (gfx1250). Guarded so we compile clean even if
// this toolchain lacks the builtins; histogram feedback tells us which path lowered.
#if defined(__HIP_DEVICE_COMPILE__)
#if __has_builtin(__builtin_amdgcn_global_load_async_to_lds_b128) && \
    __has_builtin(__builtin_amdgcn_s_wait_asynccnt)
#define USE_ASYNC_LDS 1
#endif
#endif

#if defined(USE_ASYNC_LDS)
typedef int v4i_t __attribute__((ext_vector_type(4)));
typedef __attribute__((address_space(1))) v4i_t* gv4i_p;
typedef __attribute__((address_space(3))) v4i_t* lv4i_p;

__device__ __forceinline__ void async_copy16(const float* gsrc, float* ldst) {
  // 16B per lane, global -> LDS, tracked by ASYNCcnt
  __builtin_amdgcn_global_load_async_to_lds_b128(
      (gv4i_p)gsrc, (lv4i_p)ldst, /*imm offset*/0, /*cpol*/0);
}
#endif

__launch_bounds__(256)
__global__ void vq_main(const float* __restrict__ z, const float* __restrict__ cb,
                        float* __restrict__ zq_out, float* __restrict__ idx_out,
                        float* __restrict__ partial)
{
  const int d  = threadIdx.x;           // 0..255 == column of z
  const int b0 = blockIdx.x * ROWS;

  __shared__ float s_cb[K_CB * D_DIM];  // [k][d] layout: conflict-free dequant lookup
  __shared__ float s_red[D_DIM];
#if defined(USE_ASYNC_LDS)
  __shared__ float s_z[2][STAGE_R * D_DIM];
#endif

  // Per-thread codebook row (16 floats, 64B aligned) + squared terms.
  float c[K_CB], h[K_CB];
  {
    const float4* cv = (const float4*)(cb + d * K_CB);
    float4 q0 = cv[0], q1 = cv[1], q2 = cv[2], q3 = cv[3];
    c[0]=q0.x;  c[1]=q0.y;  c[2]=q0.z;  c[3]=q0.w;
    c[4]=q1.x;  c[5]=q1.y;  c[6]=q1.z;  c[7]=q1.w;
    c[8]=q2.x;  c[9]=q2.y;  c[10]=q2.z; c[11]=q2.w;
    c[12]=q3.x; c[13]=q3.y; c[14]=q3.z; c[15]=q3.w;
#pragma unroll
    for (int k = 0; k < K_CB; ++k) {
      h[k] = c[k] * c[k];
      s_cb[k * D_DIM + d] = c[k];
    }
  }
  __syncthreads();

  float acc = 0.0f;

#if defined(USE_ASYNC_LDS)
  const int t = threadIdx.x;
  const float* gbase = z + (size_t)b0 * D_DIM;
  // Prologue: stages 0 and 1 in flight (one async instruction per wave per stage).
  async_copy16(gbase + 0 * STAGE_R * D_DIM + t * 4, &s_z[0][t * 4]);
  async_copy16(gbase + 1 * STAGE_R * D_DIM + t * 4, &s_z[1][t * 4]);

#pragma unroll 1
  for (int s = 0; s < NSTAGE; ++s) {
    if (s + 1 < NSTAGE) __builtin_amdgcn_s_wait_asynccnt(1);  // stage s landed
    else                __builtin_amdgcn_s_wait_asynccnt(0);
    __syncthreads();                                          // all waves' stage s landed
    const float* sz = s_z[s & 1];
#pragma unroll
    for (int r = 0; r < STAGE_R; ++r) {
      const int b = b0 + s * STAGE_R + r;
      const float v = sz[r * D_DIM + d];
      const float m = -2.0f * v;
      float best = fmaf(m, c[0], h[0]);
      int   bk   = 0;
#pragma unroll
      for (int k = 1; k < K_CB; ++k) {
        float sc = fmaf(m, c[k], h[k]);       // c^2 - 2 z c  (argmin == (z-c)^2)
        if (sc < best) { best = sc; bk = k; }
      }
      const float zq = s_cb[bk * D_DIM + d];  // bank-conflict-free gather
      const size_t o = (size_t)b * D_DIM + d;
      zq_out[o]  = zq;
      idx_out[o] = (float)bk;
      const float e = zq - v;
      acc = fmaf(e, e, acc);
    }
    __syncthreads();                          // everyone done reading buf (s&1)
    if (s + 2 < NSTAGE)
      async_copy16(gbase + (size_t)(s + 2) * STAGE_R * D_DIM + t * 4,
                   &s_z[s & 1][t * 4]);
  }
#else
  // Fallback: straight coalesced streaming loads.
#pragma unroll 4
  for (int r = 0; r < ROWS; ++r) {
    const int b = b0 + r;
    const float v = z[(size_t)b * D_DIM + d];
    const float m = -2.0f * v;
    float best = fmaf(m, c[0], h[0]);
    int   bk   = 0;
#pragma unroll
    for (int k = 1; k < K_CB; ++k) {
      float sc = fmaf(m, c[k], h[k]);
      if (sc < best) { best = sc; bk = k; }
    }
    const float zq = s_cb[bk * D_DIM + d];
    const size_t o = (size_t)b * D_DIM + d;
    zq_out[o]  = zq;
    idx_out[o] = (float)bk;
    const float e = zq - v;
    acc = fmaf(e, e, acc);
  }
#endif

  // Deterministic fixed-tree block reduction of the loss.
  s_red[d] = acc;
  __syncthreads();
#pragma unroll
  for (int off = 128; off > 0; off >>= 1) {
    if (d < off) s_red[d] += s_red[d + off];
    __syncthreads();
  }
  if (d == 0) partial[blockIdx.x] = s_red[0];
}

__launch_bounds__(256)
__global__ void vq_reduce(const float* __restrict__ partial, float* __restrict__ loss)
{
  __shared__ float sr[256];
  const int t = threadIdx.x;
  sr[t] = partial[t] + partial[t + 256];      // 512 partials, fixed pairing
  __syncthreads();
#pragma unroll
  for (int off = 128; off > 0; off >>= 1) {
    if (t < off) sr[t] += sr[t + off];
    __syncthreads();
  }
  if (t == 0) loss[0] = sr[0] * (1.25f / (float)B_TOT);  // (1 + 0.25) * mean over B
}

extern "C" void kernel_launch(void* const* d_in, const int* in_sizes, int n_in,
                              void* d_out, int out_size, void* d_ws, size_t ws_size,
                              hipStream_t stream) {
  (void)in_sizes; (void)n_in; (void)out_size; (void)ws_size;
  const float* z  = (const float*)d_in[0];   // (B, D) f32
  const float* cb = (const float*)d_in[1];   // (D, K) f32

  float* out     = (float*)d_out;
  float* zq_out  = out;                                  // B*D
  float* loss    = out + (size_t)B_TOT * D_DIM;          // 1
  float* idx_out = loss + 1;                             // B*D (indices as float)
  float* partial = (float*)d_ws;                         // NBLOCKS floats scratch

  vq_main<<<NBLOCKS, 256, 0, stream>>>(z, cb, zq_out, idx_out, partial);
  vq_reduce<<<1, 256, 0, stream>>>(partial, loss);
}